// Tensor_CSPNet_5781025980867
// MI455X (gfx1250) — compile-verified
//
#include <hip/hip_runtime.h>
#include <hip/hip_bf16.h>

#define S 33          // LDS row stride (pad to dodge bank conflicts)
#define EPS 1e-4f

typedef float v2f __attribute__((ext_vector_type(2)));
typedef float v8f __attribute__((ext_vector_type(8)));

// D(32x32) = op(A) * op(B), all in LDS, one wave32. Uses V_WMMA_F32_16X16X4_F32.
// Tiling: 2x2 output tiles of 16x16, K swept in 8 steps of 4.
// A operand (16x4): lanes 0-15 -> M=lane, VGPR0/1 = K0/K1 ; lanes 16-31 -> K2/K3.
// B operand (4x16): lanes 0-15 -> K row {0,1}, lanes 16-31 -> K row {2,3}, lane%16 = N.
// C/D (16x16): VGPR r -> M = r (+8 for upper lane half), lane%16 = N.
template <bool TA, bool TB>
__device__ __forceinline__ void gemm32(float* __restrict__ D,
                                       const float* __restrict__ A,
                                       const float* __restrict__ B, int lane) {
  const int half = lane >> 4;
  const int l16 = lane & 15;
#pragma unroll
  for (int mt = 0; mt < 2; ++mt) {
#pragma unroll
    for (int nt = 0; nt < 2; ++nt) {
      v8f acc = {};
      const int m = mt * 16 + l16;
      const int n = nt * 16 + l16;
#pragma unroll
      for (int kt = 0; kt < 8; ++kt) {
        const int k0 = kt * 4 + half * 2;
        v2f a, b;
        a.x = TA ? A[k0 * S + m] : A[m * S + k0];
        a.y = TA ? A[(k0 + 1) * S + m] : A[m * S + k0 + 1];
        b.x = TB ? B[n * S + k0] : B[k0 * S + n];
        b.y = TB ? B[n * S + k0 + 1] : B[(k0 + 1) * S + n];
        acc = __builtin_amdgcn_wmma_f32_16x16x4_f32(false, a, false, b,
                                                    (short)0, acc, false, false);
      }
      const int rbase = mt * 16 + half * 8;
#pragma unroll
      for (int r = 0; r < 8; ++r) D[(rbase + r) * S + n] = acc[r];
    }
  }
  __syncthreads();
}

// Parallel cyclic Jacobi eigensolver for symmetric 32x32 in LDS, one wave32.
// 16 disjoint pivot pairs per round (round-robin tournament), 31 rounds/sweep.
// On exit: diag(A) = eigenvalues, V = eigenvectors (columns).
__device__ __forceinline__ void jacobi32(float* A, float* V, int lane, int* top,
                                         int* bot, int* pp, int* qq, float* cs,
                                         float* sn) {
  for (int i = 0; i < 32; ++i) V[i * S + lane] = (i == lane) ? 1.f : 0.f;
  if (lane < 16) { top[lane] = 2 * lane; bot[lane] = 2 * lane + 1; }
  __syncthreads();
#pragma unroll 1
  for (int sweep = 0; sweep < 7; ++sweep) {
#pragma unroll 1
    for (int round = 0; round < 31; ++round) {
      if (lane < 16) {  // compute 16 independent Givens rotations
        int p = top[lane], q = bot[lane];
        if (p > q) { int t = p; p = q; q = t; }
        pp[lane] = p; qq[lane] = q;
        float apq = A[p * S + q];
        float c = 1.f, s = 0.f;
        if (__builtin_fabsf(apq) > 1e-12f) {
          float app = A[p * S + p], aqq = A[q * S + q];
          float tau = (aqq - app) / (2.f * apq);
          float t = ((tau >= 0.f) ? 1.f : -1.f) /
                    (__builtin_fabsf(tau) + __builtin_sqrtf(1.f + tau * tau));
          c = 1.f / __builtin_sqrtf(1.f + t * t);
          s = t * c;
        }
        cs[lane] = c; sn[lane] = s;
      }
      __syncthreads();
      {  // row update J^T A : lane owns column j (race-free)
        const int j = lane;
#pragma unroll 1
        for (int k = 0; k < 16; ++k) {
          int p = pp[k], q = qq[k];
          float c = cs[k], s = sn[k];
          float xv = A[p * S + j], yv = A[q * S + j];
          A[p * S + j] = c * xv - s * yv;
          A[q * S + j] = s * xv + c * yv;
        }
      }
      __syncthreads();
      {  // column update (A J) and V = V J : lane owns row i (race-free)
        const int i = lane;
#pragma unroll 1
        for (int k = 0; k < 16; ++k) {
          int p = pp[k], q = qq[k];
          float c = cs[k], s = sn[k];
          float xv = A[i * S + p], yv = A[i * S + q];
          A[i * S + p] = c * xv - s * yv;
          A[i * S + q] = s * xv + c * yv;
          float vx = V[i * S + p], vy = V[i * S + q];
          V[i * S + p] = c * vx - s * vy;
          V[i * S + q] = s * vx + c * vy;
        }
      }
      __syncthreads();
      int nt = 0, nb = 0;  // rotate tournament pairing (keep top[0] fixed)
      if (lane < 16) {
        nt = (lane == 0) ? top[0] : ((lane == 1) ? bot[0] : top[lane - 1]);
        nb = (lane == 15) ? top[15] : bot[lane + 1];
      }
      __syncthreads();
      if (lane < 16) { top[lane] = nt; bot[lane] = nb; }
      __syncthreads();
    }
  }
}

// Precompute Gneg = rm^{-1/2}, Gpos = bn^{1/2} (22x22 padded to 32x32,
// pad block stays exactly I through Jacobi since input is block-diagonal).
__global__ __launch_bounds__(32) void spd_prep(const float* __restrict__ rm,
                                               const float* __restrict__ bn,
                                               float* __restrict__ Gbuf) {
  __shared__ float M0[32 * S], V[32 * S], T[32 * S], M2[32 * S];
  __shared__ int top[16], bot[16], pp[16], qq[16];
  __shared__ float cs[16], sn[16];
  const int lane = threadIdx.x;
  const int which = blockIdx.x;
  const float* src = (which == 0) ? rm : bn;
  for (int idx = lane; idx < 1024; idx += 32) {
    int i = idx >> 5, j = idx & 31;
    float v = 0.f;
    if (i < 22 && j < 22) v = src[i * 22 + j];
    else if (i == j) v = 1.f;
    M0[i * S + j] = v;
  }
  __syncthreads();
  jacobi32(M0, V, lane, top, bot, pp, qq, cs, sn);
  {
    float wv = M0[lane * S + lane];
    float f = (which == 0) ? (1.f / __builtin_sqrtf(fmaxf(wv, 1e-12f)))
                           : __builtin_sqrtf(fmaxf(wv, 0.f));
    for (int i = 0; i < 32; ++i) T[i * S + lane] = V[i * S + lane] * f;
  }
  __syncthreads();
  gemm32<false, true>(M2, T, V, lane);  // G = V diag(f(w)) V^T
  for (int idx = lane; idx < 1024; idx += 32)
    Gbuf[which * 1024 + idx] = M2[(idx >> 5) * S + (idx & 31)];
}

// One wave32 per (b,h) SPD matrix. All GEMMs via f32 WMMA; 2 Jacobi eigensolves
// (ReEig is fused with LogEig: same eigenvectors -> log(max(w,EPS))).
__global__ __launch_bounds__(32) void spd_main(
    const float* __restrict__ x, const float* __restrict__ W1,
    const float* __restrict__ W2, const float* __restrict__ conv_w,
    const float* __restrict__ Gbuf, float* __restrict__ feat) {
  __shared__ float M0[32 * S], M1[32 * S], M2[32 * S], V[32 * S], Wa[32 * S];
  __shared__ int top[16], bot[16], pp[16], qq[16];
  __shared__ float cs[16], sn[16];
  const int lane = threadIdx.x;
  const int bid = blockIdx.x;  // 0 .. 512*27-1
  const int b = bid / 27;
  const int h = bid % 27;
  const int wwin = h / 9, band = h % 9;
  const float* cwbase = conv_w + wwin * 4356 + band * 484;

  if (lane < 48) __builtin_prefetch(cwbase + (size_t)lane * 13068, 0, 1);

  // Wa = W1[h] (32x32)
  const float* w1h = W1 + (size_t)h * 1024;
  for (int idx = lane; idx < 1024; idx += 32)
    Wa[(idx >> 5) * S + (idx & 31)] = w1h[idx];
  // M0 = X padded to 32x32 with ones on the pad diagonal (matches reference)
  const float* xbh = x + (size_t)bid * 484;
  for (int idx = lane; idx < 1024; idx += 32) {
    int i = idx >> 5, j = idx & 31;
    float v = 0.f;
    if (i < 22 && j < 22) v = xbh[i * 22 + j];
    else if (i == j) v = 1.f;
    M0[i * S + j] = v;
  }
  __syncthreads();
  gemm32<true, false>(M1, Wa, M0, lane);   // M1 = W1^T X
  gemm32<false, false>(M2, M1, Wa, lane);  // M2 = W1^T X W1

  jacobi32(M2, V, lane, top, bot, pp, qq, cs, sn);  // ReEig #1
  {
    float f = fmaxf(M2[lane * S + lane], EPS);
    for (int i = 0; i < 32; ++i) M0[i * S + lane] = V[i * S + lane] * f;
  }
  __syncthreads();
  gemm32<false, true>(M1, M0, V, lane);    // M1 = V diag(max(w,EPS)) V^T

  // Wa = W2[h] (32x22) zero-padded to 32x32
  const float* w2h = W2 + (size_t)h * (32 * 22);
  for (int idx = lane; idx < 1024; idx += 32) {
    int i = idx >> 5, j = idx & 31;
    Wa[i * S + j] = (j < 22) ? w2h[i * 22 + j] : 0.f;
  }
  __syncthreads();
  gemm32<true, false>(M2, Wa, M1, lane);   // W2^T P
  gemm32<false, false>(M0, M2, Wa, lane);  // M0 = W2^T P W2 (22x22 block)

  // congruence with rm^{-1/2}
  for (int idx = lane; idx < 1024; idx += 32)
    Wa[(idx >> 5) * S + (idx & 31)] = Gbuf[idx];
  __syncthreads();
  gemm32<false, false>(M1, Wa, M0, lane);
  gemm32<false, false>(M2, M1, Wa, lane);
  // congruence with bn^{1/2}
  for (int idx = lane; idx < 1024; idx += 32)
    Wa[(idx >> 5) * S + (idx & 31)] = Gbuf[1024 + idx];
  __syncthreads();
  gemm32<false, false>(M0, Wa, M2, lane);
  gemm32<false, false>(M1, M0, Wa, lane);

  // pad diagonal -> 1 so the padded block has eigenvalue 1 (log -> 0)
  if (lane >= 22) M1[lane * S + lane] = 1.f;
  __syncthreads();
  jacobi32(M1, V, lane, top, bot, pp, qq, cs, sn);  // fused ReEig + LogEig
  {
    float f = __builtin_logf(fmaxf(M1[lane * S + lane], EPS));
    for (int i = 0; i < 32; ++i) M0[i * S + lane] = V[i * S + lane] * f;
  }
  __syncthreads();
  gemm32<false, true>(M2, M0, V, lane);    // M2 = logm(P), valid 22x22 block

  // conv feature accumulation: lane covers c in {lane, lane+32}
  for (int c = lane; c < 48; c += 32) {
    const float* cw = cwbase + (size_t)c * 13068;
    float sacc = 0.f;
    for (int i = 0; i < 22; ++i) {
      const float* mrow = &M2[i * S];
      const float* crow = cw + i * 22;
#pragma unroll 1
      for (int j = 0; j < 22; ++j) sacc += mrow[j] * crow[j];
    }
    __hip_atomic_fetch_add(&feat[b * 48 + c], sacc, __ATOMIC_RELAXED,
                           __HIP_MEMORY_SCOPE_AGENT);
  }
}

__global__ void zero_feat(float* __restrict__ feat, int n) {
  int i = blockIdx.x * blockDim.x + threadIdx.x;
  if (i < n) feat[i] = 0.f;
}

__global__ void fc_kernel(const float* __restrict__ feat,
                          const float* __restrict__ conv_b,
                          const float* __restrict__ fc_w,
                          const float* __restrict__ fc_b,
                          float* __restrict__ out) {
  int i = blockIdx.x * blockDim.x + threadIdx.x;  // 0 .. 512*4-1
  if (i >= 512 * 4) return;
  int b = i >> 2, k = i & 3;
  float s = fc_b[k];
#pragma unroll 1
  for (int c = 0; c < 48; ++c)
    s += (feat[b * 48 + c] + conv_b[c]) * fc_w[k * 48 + c];
  out[i] = s;
}

extern "C" void kernel_launch(void* const* d_in, const int* in_sizes, int n_in,
                              void* d_out, int out_size, void* d_ws,
                              size_t ws_size, hipStream_t stream) {
  (void)in_sizes; (void)n_in; (void)out_size; (void)ws_size;
  const float* x       = (const float*)d_in[0];  // (512,3,9,22,22)
  const float* W1      = (const float*)d_in[1];  // (27,32,32)
  const float* W2      = (const float*)d_in[2];  // (27,32,22)
  const float* rm      = (const float*)d_in[3];  // (22,22)
  const float* bn      = (const float*)d_in[4];  // (22,22)
  const float* conv_w  = (const float*)d_in[5];  // (48,1,3,4356)
  const float* conv_b  = (const float*)d_in[6];  // (48,)
  const float* fc_w    = (const float*)d_in[7];  // (4,48)
  const float* fc_b    = (const float*)d_in[8];  // (4,)
  float* out = (float*)d_out;                    // (512,4)
  float* ws = (float*)d_ws;
  float* Gbuf = ws;            // 2 * 1024 floats: Gneg_pad, Gpos_pad
  float* feat = ws + 2048;     // 512 * 48 floats

  zero_feat<<<(512 * 48 + 255) / 256, 256, 0, stream>>>(feat, 512 * 48);
  spd_prep<<<2, 32, 0, stream>>>(rm, bn, Gbuf);
  spd_main<<<512 * 27, 32, 0, stream>>>(x, W1, W2, conv_w, Gbuf, feat);
  fc_kernel<<<(512 * 4 + 255) / 256, 256, 0, stream>>>(feat, conv_b, fc_w,
                                                       fc_b, out);
}